// TemporalGraphEncoder_49709951484667
// MI455X (gfx1250) — compile-verified
//
#include <hip/hip_runtime.h>
#include <hip/hip_bf16.h>

typedef __attribute__((ext_vector_type(2))) float v2f;
typedef __attribute__((ext_vector_type(8))) float v8f;

// ---------------------------------------------------------------------------
// Init: smax = -inf, denom = 0
// ---------------------------------------------------------------------------
__global__ void tgn_init_kernel(float* __restrict__ smax, float* __restrict__ denom, int n2) {
    int i = blockIdx.x * blockDim.x + threadIdx.x;
    if (i < n2) {
        ((unsigned int*)smax)[i] = 0xFF800000u;  // -inf
        denom[i] = 0.0f;
    }
}

// ---------------------------------------------------------------------------
// Fused gather + Q/K/V/skip GEMM using native f32 WMMA (16x16x4).
// Block = 128 threads = 4 waves, one 16-row M tile per block.
// Q/K/V tiles (ld=128) and skip tiles (ld=64) are split into separate loops
// so all weight-load offsets are compile-time constants (fold into IOFFSET).
// ---------------------------------------------------------------------------
__global__ __launch_bounds__(128)
void tgn_gemm_kernel(const int* __restrict__ n_id,
                     const float* __restrict__ mem_t,   // [NUM_NODES,64]
                     const float* __restrict__ emb_t,   // [NUM_NODES,64]
                     const float* __restrict__ Wq, const float* __restrict__ bq,
                     const float* __restrict__ Wk, const float* __restrict__ bk,
                     const float* __restrict__ Wv, const float* __restrict__ bv,
                     const float* __restrict__ Ws, const float* __restrict__ bs,
                     float* __restrict__ q, float* __restrict__ k,
                     float* __restrict__ v, float* __restrict__ out,
                     int N) {
    __shared__ float xs[16][132];   // 16x128 tile, padded to avoid bank conflicts

    const int row0 = blockIdx.x * 16;
    const int tid  = threadIdx.x;

    // ---- cooperative gather: 8 threads per row, 16 floats (4x float4) each
    {
        int r    = tid >> 3;        // 0..15
        int cseg = (tid & 7) * 16;  // 0,16,...,112
        int gr   = row0 + r;
        float4* dst4 = (float4*)&xs[r][cseg];   // 16B aligned
        if (gr < N) {
            size_t nid = (size_t)n_id[gr];
            const float4* src4 = (cseg < 64)
                ? (const float4*)(emb_t + nid * 64 + cseg)
                : (const float4*)(mem_t + nid * 64 + (cseg - 64));
#pragma unroll
            for (int i = 0; i < 4; ++i) dst4[i] = src4[i];
        } else {
            float4 z = make_float4(0.f, 0.f, 0.f, 0.f);
#pragma unroll
            for (int i = 0; i < 4; ++i) dst4[i] = z;
        }
    }
    __syncthreads();

    const int wave = tid >> 5;
    const int lane = tid & 31;
    const int m    = lane & 15;     // A-row / B,C,D-column lane index
    const int kh   = lane >> 4;     // K half: 0 -> K{0,1}, 1 -> K{2,3}

    const float* xb = &xs[m][2 * kh];   // A lane base (8B aligned)

    // ---- Q/K/V tiles: 24 tiles of 16 columns, ld = 128 (constant)
    for (int ct = wave; ct < 24; ct += 4) {
        const float* wptr; const float* bptr; float* dptr; int cb;
        if (ct < 8)       { wptr = Wq; bptr = bq; dptr = q; cb = ct * 16; }
        else if (ct < 16) { wptr = Wk; bptr = bk; dptr = k; cb = (ct - 8)  * 16; }
        else              { wptr = Wv; bptr = bv; dptr = v; cb = (ct - 16) * 16; }

        const float bias = bptr[cb + m];
        v8f acc;
#pragma unroll
        for (int j = 0; j < 8; ++j) acc[j] = bias;

        // lane weight base: row (2*kh), col (cb+m); per-step offsets constant
        const float* wb = wptr + (size_t)(2 * kh) * 128 + cb + m;
#pragma unroll
        for (int kk = 0; kk < 32; ++kk) {
            v2f a = *(const v2f*)(xb + kk * 4);                 // K = 4kk+2kh, +1
            v2f b;
            b.x = wb[kk * 512];        // W[4kk+2kh  ][cb+m]
            b.y = wb[kk * 512 + 128];  // W[4kk+2kh+1][cb+m]
            acc = __builtin_amdgcn_wmma_f32_16x16x4_f32(
                false, a, false, b, (short)0, acc, false, false);
        }

        // D layout: VGPR j -> row (j + 8*kh), col (cb + m)
#pragma unroll
        for (int j = 0; j < 8; ++j) {
            int r = row0 + j + 8 * kh;
            if (r < N) dptr[(size_t)r * 128 + cb + m] = acc[j];
        }
    }

    // ---- skip tile: 4 tiles of 16 columns, ld = 64 (constant), one per wave
    {
        const int cb = wave * 16;
        const float bias = bs[cb + m];
        v8f acc;
#pragma unroll
        for (int j = 0; j < 8; ++j) acc[j] = bias;

        const float* wb = Ws + (size_t)(2 * kh) * 64 + cb + m;
#pragma unroll
        for (int kk = 0; kk < 32; ++kk) {
            v2f a = *(const v2f*)(xb + kk * 4);
            v2f b;
            b.x = wb[kk * 256];
            b.y = wb[kk * 256 + 64];
            acc = __builtin_amdgcn_wmma_f32_16x16x4_f32(
                false, a, false, b, (short)0, acc, false, false);
        }

#pragma unroll
        for (int j = 0; j < 8; ++j) {
            int r = row0 + j + 8 * kh;
            if (r < N) out[(size_t)r * 64 + cb + m] = acc[j];
        }
    }
}

// ---------------------------------------------------------------------------
// Per (edge, head): score = dot(q[dst,h], k[src,h]) / 8; atomic float max.
// ---------------------------------------------------------------------------
__global__ void tgn_score_kernel(const int* __restrict__ ei,
                                 const float* __restrict__ q,
                                 const float* __restrict__ k,
                                 float* __restrict__ smax,
                                 float* __restrict__ score,
                                 int E) {
    int idx = blockIdx.x * blockDim.x + threadIdx.x;
    if (idx >= 2 * E) return;
    int e = idx >> 1, h = idx & 1;
    int src = ei[e];
    int dst = ei[E + e];
    const float4* qp = (const float4*)(q + (size_t)dst * 128 + h * 64);
    const float4* kp = (const float4*)(k + (size_t)src * 128 + h * 64);
    float s = 0.f;
#pragma unroll
    for (int i = 0; i < 16; ++i) {
        float4 a = qp[i], b = kp[i];
        s += a.x * b.x + a.y * b.y + a.z * b.z + a.w * b.w;
    }
    s *= 0.125f;  // 1/sqrt(64)
    score[idx] = s;
    // float atomic max via signed/unsigned int atomics (init bits = -inf)
    float* addr = smax + (size_t)dst * 2 + h;
    if (s >= 0.f) atomicMax((int*)addr, __float_as_int(s));
    else          atomicMin((unsigned int*)addr, __float_as_uint(s));
}

// ---------------------------------------------------------------------------
// Per (edge, head): exps = exp(score - smax[dst]); denom[dst] += exps.
// ---------------------------------------------------------------------------
__global__ void tgn_expsum_kernel(const int* __restrict__ ei,
                                  const float* __restrict__ smax,
                                  float* __restrict__ score,
                                  float* __restrict__ denom,
                                  int E) {
    int idx = blockIdx.x * blockDim.x + threadIdx.x;
    if (idx >= 2 * E) return;
    int e = idx >> 1, h = idx & 1;
    int dst = ei[E + e];
    float mx = smax[(size_t)dst * 2 + h];
    float ex = __expf(score[idx] - mx);
    score[idx] = ex;  // overwrite in place
    atomicAdd(denom + (size_t)dst * 2 + h, ex);
}

// ---------------------------------------------------------------------------
// Per edge: out[dst] += 0.5*(alpha0*v[src,0] + alpha1*v[src,1])
// ---------------------------------------------------------------------------
__global__ void tgn_agg_kernel(const int* __restrict__ ei,
                               const float* __restrict__ v,
                               const float* __restrict__ score,
                               const float* __restrict__ denom,
                               float* __restrict__ out,
                               int E) {
    int e = blockIdx.x * blockDim.x + threadIdx.x;
    if (e >= E) return;
    int src = ei[e];
    int dst = ei[E + e];
    float a0 = 0.5f * score[(size_t)e * 2 + 0] / denom[(size_t)dst * 2 + 0];
    float a1 = 0.5f * score[(size_t)e * 2 + 1] / denom[(size_t)dst * 2 + 1];
    const float4* v0 = (const float4*)(v + (size_t)src * 128);
    const float4* v1 = v0 + 16;
    float* o = out + (size_t)dst * 64;
#pragma unroll
    for (int i = 0; i < 16; ++i) {
        float4 x0 = v0[i], x1 = v1[i];
        atomicAdd(o + 4 * i + 0, a0 * x0.x + a1 * x1.x);
        atomicAdd(o + 4 * i + 1, a0 * x0.y + a1 * x1.y);
        atomicAdd(o + 4 * i + 2, a0 * x0.z + a1 * x1.z);
        atomicAdd(o + 4 * i + 3, a0 * x0.w + a1 * x1.w);
    }
}

// ---------------------------------------------------------------------------
extern "C" void kernel_launch(void* const* d_in, const int* in_sizes, int n_in,
                              void* d_out, int out_size, void* d_ws, size_t ws_size,
                              hipStream_t stream) {
    const int*   n_id = (const int*)  d_in[0];
    const int*   ei   = (const int*)  d_in[1];   // [2, E]
    const float* memt = (const float*)d_in[2];
    const float* embt = (const float*)d_in[3];
    const float* Wq = (const float*)d_in[4];  const float* bq = (const float*)d_in[5];
    const float* Wk = (const float*)d_in[6];  const float* bk = (const float*)d_in[7];
    const float* Wv = (const float*)d_in[8];  const float* bv = (const float*)d_in[9];
    const float* Ws = (const float*)d_in[10]; const float* bs = (const float*)d_in[11];

    const int N = in_sizes[0];
    const int E = in_sizes[1] / 2;

    float* ws    = (float*)d_ws;
    float* q     = ws;                       // N*128
    float* k     = q + (size_t)N * 128;      // N*128
    float* v     = k + (size_t)N * 128;      // N*128
    float* smax  = v + (size_t)N * 128;      // N*2
    float* denom = smax + (size_t)N * 2;     // N*2
    float* score = denom + (size_t)N * 2;    // E*2
    float* out   = (float*)d_out;            // N*64 (skip written, then atomic adds)

    tgn_init_kernel<<<(2 * N + 255) / 256, 256, 0, stream>>>(smax, denom, 2 * N);
    tgn_gemm_kernel<<<(N + 15) / 16, 128, 0, stream>>>(
        n_id, memt, embt, Wq, bq, Wk, bk, Wv, bv, Ws, bs, q, k, v, out, N);
    tgn_score_kernel<<<(2 * E + 255) / 256, 256, 0, stream>>>(ei, q, k, smax, score, E);
    tgn_expsum_kernel<<<(2 * E + 255) / 256, 256, 0, stream>>>(ei, smax, score, denom, E);
    tgn_agg_kernel<<<(E + 255) / 256, 256, 0, stream>>>(ei, v, score, denom, out, E);
}